// NER_60945585930503
// MI455X (gfx1250) — compile-verified
//
#include <hip/hip_runtime.h>

typedef float v2f __attribute__((ext_vector_type(2)));
typedef float v8f __attribute__((ext_vector_type(8)));

#define B_   8
#define L_   512
#define H_   768
#define T_   9
#define NQ_  (H_ * T_)   // 6912 output columns of each projection

// ---------------------------------------------------------------------------
// Kernel 0: RoPE sin/cos tables. sin_t[l*32+j] = sin(l * 10000^(-j/32))
// ---------------------------------------------------------------------------
__global__ void rope_tables(float* __restrict__ sin_t, float* __restrict__ cos_t) {
    int idx = blockIdx.x * blockDim.x + threadIdx.x;   // 0 .. 16383
    int l = idx >> 5;
    int j = idx & 31;
    float inv = powf(10000.0f, -(float)j / 32.0f);
    float ang = (float)l * inv;
    sin_t[idx] = sinf(ang);
    cos_t[idx] = cosf(ang);
}

// ---------------------------------------------------------------------------
// Kernel 1: projection GEMM (hidden @ W + b) with fused interleaved RoPE.
// One wave computes a 32(M) x 64(N) tile with V_WMMA_F32_16X16X4_F32:
// two A fragments share four B fragments -> 8 wmma per 10 vmem per K-step.
// Output layout: (b, t, l, 768) so the score GEMM reads contiguous rows.
// blockIdx.z: 0 -> q (Wq,bq), 1 -> k (Wk,bk)
// ---------------------------------------------------------------------------
__global__ void __launch_bounds__(32)
proj_rope(const float* __restrict__ hidden,                    // (B*L, 768)
          const float* __restrict__ Wq, const float* __restrict__ bq,
          const float* __restrict__ Wk, const float* __restrict__ bk,
          const float* __restrict__ sin_t, const float* __restrict__ cos_t,
          float* __restrict__ qrot, float* __restrict__ krot)
{
    const int lane  = threadIdx.x;
    const int half  = lane >> 4;      // K-half for A/B fragments, M-half for D
    const int lm    = lane & 15;
    const int nBase = blockIdx.x * 64;
    const int mBase = blockIdx.y * 32;
    const bool isK  = (blockIdx.z != 0);

    const float* W    = isK ? Wk : Wq;
    const float* bias = isK ? bk : bq;
    float*       dst  = isK ? krot : qrot;

    // C initialized with the bias (bias depends only on N == lane column)
    v8f acc[2][4];
#pragma unroll
    for (int u = 0; u < 4; ++u) {
        float bv = bias[nBase + u * 16 + lm];
#pragma unroll
        for (int v = 0; v < 8; ++v) { acc[0][u][v] = bv; acc[1][u][v] = bv; }
    }

    // A fragment rows: lane lm -> M = mBase + w*16 + lm;
    // VGPR pair holds K = kk + 2*half, +1
    const float* hrow0 = hidden + (size_t)(mBase + lm) * H_;
    const float* hrow1 = hidden + (size_t)(mBase + 16 + lm) * H_;

    for (int kk = 0; kk < H_; kk += 4) {
        const int krow = kk + 2 * half;
        v2f a0 = *(const v2f*)(hrow0 + krow);
        v2f a1 = *(const v2f*)(hrow1 + krow);
        const float* w0 = W + (size_t)krow * NQ_ + nBase + lm;
        const float* w1 = w0 + NQ_;
#pragma unroll
        for (int u = 0; u < 4; ++u) {
            v2f b;
            b.x = w0[u * 16];
            b.y = w1[u * 16];
            acc[0][u] = __builtin_amdgcn_wmma_f32_16x16x4_f32(
                false, a0, false, b, (short)0, acc[0][u], false, false);
            acc[1][u] = __builtin_amdgcn_wmma_f32_16x16x4_f32(
                false, a1, false, b, (short)0, acc[1][u], false, false);
        }
    }

    // Fused interleaved RoPE epilogue.
    // D layout: element v of lane -> M = w*16 + v + 8*half, N = nBase+u*16+lm.
    // Adjacent columns live in adjacent lanes -> pair partner via shfl_xor(1).
    const int bIdx  = mBase >> 9;          // batch (tiles never straddle batches)
    const int lBase = mBase & (L_ - 1);
#pragma unroll
    for (int u = 0; u < 4; ++u) {
        const int n    = nBase + u * 16 + lm;
        const int t    = n / H_;
        const int cc   = n - t * H_;       // column within 768
        const int dd   = cc & 63;          // position within head
        const int j    = dd >> 1;          // rotation pair index
        const bool evn = (dd & 1) == 0;
        float* base = dst + ((size_t)(bIdx * T_ + t) * L_) * H_ + cc;
#pragma unroll
        for (int w = 0; w < 2; ++w) {
#pragma unroll
            for (int v = 0; v < 8; ++v) {
                float x = acc[w][u][v];
                float p = __shfl_xor(x, 1, 32);          // partner column value
                int lpos = lBase + w * 16 + v + 8 * half; // sequence position
                float s = sin_t[lpos * 32 + j];
                float c = cos_t[lpos * 32 + j];
                float r = evn ? (x * c - p * s) : (x * c + p * s);
                base[(size_t)lpos * H_] = r;
            }
        }
    }
}

// ---------------------------------------------------------------------------
// Kernel 2: score[b,i,j,t] = (scale/HEADS) * dot(qrot[b,t,i,:], krot[b,t,j,:])
// One wave computes a 32(i) x 64(j) tile for one (b,t). Both A and B
// fragments are contiguous float2 loads thanks to the (b,t,l,768) layout.
// ---------------------------------------------------------------------------
__global__ void __launch_bounds__(32)
score_gemm(const float* __restrict__ qrot, const float* __restrict__ krot,
           float* __restrict__ out)
{
    const int lane  = threadIdx.x;
    const int half  = lane >> 4;
    const int lm    = lane & 15;
    const int nBase = blockIdx.x * 64;   // j
    const int mBase = blockIdx.y * 32;   // i
    const int bt    = blockIdx.z;        // b*T + t
    const int b     = bt / T_;
    const int t     = bt - b * T_;

    const float* Q = qrot + (size_t)bt * L_ * H_;
    const float* K = krot + (size_t)bt * L_ * H_;

    v8f acc[2][4] = {};
    const float* qrow0 = Q + (size_t)(mBase + lm) * H_;
    const float* qrow1 = Q + (size_t)(mBase + 16 + lm) * H_;
    const float* krow0 = K + (size_t)(nBase + lm) * H_;

    for (int kk = 0; kk < H_; kk += 4) {
        const int krow = kk + 2 * half;
        v2f a0 = *(const v2f*)(qrow0 + krow);
        v2f a1 = *(const v2f*)(qrow1 + krow);
#pragma unroll
        for (int u = 0; u < 4; ++u) {
            v2f bf = *(const v2f*)(krow0 + (size_t)(u * 16) * H_ + krow);
            acc[0][u] = __builtin_amdgcn_wmma_f32_16x16x4_f32(
                false, a0, false, bf, (short)0, acc[0][u], false, false);
            acc[1][u] = __builtin_amdgcn_wmma_f32_16x16x4_f32(
                false, a1, false, bf, (short)0, acc[1][u], false, false);
        }
    }

    const float sc = 0.125f / 12.0f;     // (H/12)^-0.5 / HEADS
#pragma unroll
    for (int u = 0; u < 4; ++u) {
        const int j = nBase + u * 16 + lm;
#pragma unroll
        for (int w = 0; w < 2; ++w) {
#pragma unroll
            for (int v = 0; v < 8; ++v) {
                const int i = mBase + w * 16 + v + 8 * half;
                out[(((size_t)(b * L_ + i) * L_) + j) * T_ + t] = acc[w][u][v] * sc;
            }
        }
    }
}

// ---------------------------------------------------------------------------
extern "C" void kernel_launch(void* const* d_in, const int* in_sizes, int n_in,
                              void* d_out, int out_size, void* d_ws, size_t ws_size,
                              hipStream_t stream) {
    const float* hidden = (const float*)d_in[0];
    const float* Wq     = (const float*)d_in[1];
    const float* bq     = (const float*)d_in[2];
    const float* Wk     = (const float*)d_in[3];
    const float* bk     = (const float*)d_in[4];
    float* out = (float*)d_out;

    float* ws    = (float*)d_ws;
    float* sin_t = ws;                       // 512*32
    float* cos_t = ws + 16384;               // 512*32
    float* qrot  = ws + 32768;               // B*T*L*H floats
    float* krot  = qrot + (size_t)B_ * T_ * L_ * H_;

    rope_tables<<<64, 256, 0, stream>>>(sin_t, cos_t);

    proj_rope<<<dim3(NQ_ / 64, (B_ * L_) / 32, 2), dim3(32), 0, stream>>>(
        hidden, Wq, bq, Wk, bk, sin_t, cos_t, qrot, krot);

    score_gemm<<<dim3(L_ / 64, L_ / 32, B_ * T_), dim3(32), 0, stream>>>(
        qrot, krot, out);
}